// PathPredNet_45535243272619
// MI455X (gfx1250) — compile-verified
//
#include <hip/hip_runtime.h>

// ---------------------------------------------------------------------------
// PathPredNet fused kernel for gfx1250 (MI455X, wave32, WMMA).
//
//   x  = concat(paths[n], Z_pat[n], actors[u[n]], Z_act[u[n]])      [N,320]
//   h1 = relu(GN1(x @ w1^T))                                        [N,160]
//   h2 = GN2(h1 @ w2^T)                                             [N,160]
//   h  = relu(h2 + x @ wt^T)                                        [N,160]
//   y  = h @ wh^T + bh                                              [N,1]
//
// 102.4 GFLOP vs 260 MB HBM traffic (~395 FLOP/B) -> compute bound.
// All GEMMs run on V_WMMA_F32_16X16X4_F32 (fp32-exact matrix pipe).
// Block = 320 threads (10 wave32), 64 rows/block, everything fused in LDS.
// ---------------------------------------------------------------------------

typedef float v2f __attribute__((ext_vector_type(2)));
typedef float v8f __attribute__((ext_vector_type(8)));

namespace {
constexpr int   kNActor  = 128;
constexpr int   kNMap    = 128;
constexpr int   kActC    = 32;
constexpr int   kPatC    = 32;
constexpr int   kSize    = 320;   // concat width (K of stage 1)
constexpr int   kHid     = 160;   // hidden width (K of stage 2)
constexpr float kEps     = 1e-5f;

constexpr int   kRows    = 64;    // path rows per block (4 x 16-row slabs)
constexpr int   kSlabs   = 4;
constexpr int   kThreads = 320;   // 10 wave32 waves

// LDS row strides chosen == 4 (mod 32) so a 16-row x 4-dword WMMA fragment
// read (b64 per lane) touches 64 distinct banks; even -> 8B aligned rows.
constexpr int   kXStr    = 324;
constexpr int   kHStr    = 164;

constexpr int    kXsF       = kRows * kXStr;                      // 20736
constexpr int    kHF        = kRows * kHStr;                      // 10496
constexpr int    kSmemF     = kXsF + 2 * kHF + 320 + 320 + 64 + 64; // 42496
constexpr size_t kSmemBytes = (size_t)kSmemF * sizeof(float);     // 169984 (< 320KB/WGP)
} // namespace

__device__ __forceinline__ v8f wmma_f32_k4(v2f a, v2f b, v8f c) {
  // D = A(16x4) * B(4x16) + C(16x16), fp32 matrix pipe.
  return __builtin_amdgcn_wmma_f32_16x16x4_f32(false, a, false, b,
                                               (short)0, c, false, false);
}

__launch_bounds__(kThreads, 1)
__global__ void pathpred_kernel(const float* __restrict__ actors,
                                const float* __restrict__ paths,
                                const float* __restrict__ Zact,
                                const float* __restrict__ Zpat,
                                const int*   __restrict__ u,
                                const float* __restrict__ w1,
                                const float* __restrict__ w2,
                                const float* __restrict__ wt,
                                const float* __restrict__ g1w,
                                const float* __restrict__ g1b,
                                const float* __restrict__ g2w,
                                const float* __restrict__ g2b,
                                const float* __restrict__ wh,
                                const float* __restrict__ bh,
                                float* __restrict__ out) {
  extern __shared__ float smem[];
  float* xs  = smem;               // [64][324] gathered concat x (fp32)
  float* h1s = xs  + kXsF;         // [64][164] h1 (then normalized h1)
  float* hts = h1s + kHF;          // [64][164] transform branch x@wt^T
  float* h2s = xs;                 // [64][164] aliases xs (dead after stage 1)
  float* sA  = hts + kHF;          // [5][64] partial sums / dot partials
  float* sB  = sA + 320;           // [5][64] partial sum-of-squares
  float* rmu = sB + 320;           // [64] per-row mean
  float* rrs = rmu + 64;           // [64] per-row rsqrt(var+eps)

  const int t     = threadIdx.x;
  const int base  = blockIdx.x * kRows;   // N_PATH = 400000 = 6250 * 64 exactly
  const int wave  = t >> 5;
  const int lane  = t & 31;
  const int laneN = lane & 15;
  const int half  = lane >> 4;            // WMMA half-wave (K/M split)

  // ------------------------------------------------------------------
  // 1) Gather + concat x tile into LDS. Thread t owns column t for all
  //    64 rows; segment boundaries (128/160/288) are multiples of 32 so
  //    waves never diverge. u[base+r] is uniform -> scalar load.
  // ------------------------------------------------------------------
  {
    const int col = t;
    for (int r = 0; r < kRows; ++r) {
      const int n = base + r;
      float v;
      if (col < kNMap) {
        v = paths[(size_t)n * kNMap + col];
      } else if (col < kNMap + kPatC) {
        v = Zpat[(size_t)n * kPatC + (col - kNMap)];
      } else {
        const int a = u[n];
        if (col < kNMap + kPatC + kNActor)
          v = actors[(size_t)a * kNActor + (col - (kNMap + kPatC))];
        else
          v = Zact[(size_t)a * kActC + (col - (kNMap + kPatC + kNActor))];
      }
      xs[r * kXStr + col] = v;
    }
  }
  __syncthreads();

  // ------------------------------------------------------------------
  // 2) Stage-1 GEMM: [64,320] x [320,320]^T where the 320 outputs are
  //    the fused (w1 | wt) rows. Wave w owns output cols 32w..32w+31
  //    (two 16-wide N-tiles), all 4 M-slabs. B fragment is reused across
  //    the 4 slabs -> 4x less L2 weight traffic. 8 independent WMMAs
  //    per k-step hide the matrix-pipe latency.
  // ------------------------------------------------------------------
  {
    v8f acc[2][kSlabs];
#pragma unroll
    for (int n = 0; n < 2; ++n)
#pragma unroll
      for (int s = 0; s < kSlabs; ++s)
#pragma unroll
        for (int i = 0; i < 8; ++i) acc[n][s][i] = 0.0f;

    const float* bptr[2];
#pragma unroll
    for (int n = 0; n < 2; ++n) {
      const int jj = wave * 32 + n * 16 + laneN;   // fused output column
      const float* W = (jj < kHid) ? w1 : wt;      // uniform per (wave,n)
      const int wr   = (jj < kHid) ? jj : jj - kHid;
      bptr[n] = W + (size_t)wr * kSize;
    }

    for (int k = 0; k < kSize; k += 4) {
      const int kk = k + 2 * half;
      const v2f b0 = *(const v2f*)(bptr[0] + kk);   // global_load_b64 (L2 hit)
      const v2f b1 = *(const v2f*)(bptr[1] + kk);
      v2f a[kSlabs];
#pragma unroll
      for (int s = 0; s < kSlabs; ++s)              // ds_load_b64, bank-clean
        a[s] = *(const v2f*)(&xs[(s * 16 + laneN) * kXStr + kk]);
#pragma unroll
      for (int s = 0; s < kSlabs; ++s) {
        acc[0][s] = wmma_f32_k4(a[s], b0, acc[0][s]);
        acc[1][s] = wmma_f32_k4(a[s], b1, acc[1][s]);
      }
    }

    // Epilogue: C/D layout -> element (M = v + 8*half, N = laneN).
#pragma unroll
    for (int n = 0; n < 2; ++n) {
      const int jj   = wave * 32 + n * 16 + laneN;
      float* dst     = (jj < kHid) ? h1s : hts;     // uniform per (wave,n)
      const int jc   = (jj < kHid) ? jj : jj - kHid;
#pragma unroll
      for (int s = 0; s < kSlabs; ++s)
#pragma unroll
        for (int v = 0; v < 8; ++v) {
          const int row = s * 16 + v + 8 * half;
          dst[row * kHStr + jc] = acc[n][s][v];
        }
    }
  }
  __syncthreads();

  // ------------------------------------------------------------------
  // 3) GN1 + ReLU over h1 rows (160 channels). 5 chunk-threads per row.
  // ------------------------------------------------------------------
  {
    const int row = t & 63, chunk = t >> 6;         // chunk in [0,5)
    float s = 0.0f, q = 0.0f;
    for (int i = 0; i < 32; ++i) {
      const float v = h1s[row * kHStr + chunk * 32 + i];
      s += v; q += v * v;
    }
    sA[chunk * 64 + row] = s;
    sB[chunk * 64 + row] = q;
  }
  __syncthreads();
  if (t < 64) {
    float s = 0.0f, q = 0.0f;
    for (int c = 0; c < 5; ++c) { s += sA[c * 64 + t]; q += sB[c * 64 + t]; }
    const float mu  = s * (1.0f / kHid);
    const float var = q * (1.0f / kHid) - mu * mu;
    rmu[t] = mu;
    rrs[t] = rsqrtf(var + kEps);
  }
  __syncthreads();
  {
    const int c = t % kHid, r0 = t / kHid;          // r0 in {0,1}
    const float gw = g1w[c], gb = g1b[c];
    for (int rr = 0; rr < kRows / 2; ++rr) {
      const int row = rr * 2 + r0;
      float v = h1s[row * kHStr + c];
      v = (v - rmu[row]) * rrs[row] * gw + gb;
      h1s[row * kHStr + c] = fmaxf(v, 0.0f);
    }
  }
  __syncthreads();

  // ------------------------------------------------------------------
  // 4) Stage-2 GEMM: h2 = h1n @ w2^T. Wave w owns N-tile 16w, K=160.
  //    h2 lands in the (now dead) xs region.
  // ------------------------------------------------------------------
  {
    v8f acc[kSlabs];
#pragma unroll
    for (int s = 0; s < kSlabs; ++s)
#pragma unroll
      for (int i = 0; i < 8; ++i) acc[s][i] = 0.0f;

    const float* bptr = w2 + (size_t)(wave * 16 + laneN) * kHid;
    for (int k = 0; k < kHid; k += 4) {
      const int kk = k + 2 * half;
      const v2f b = *(const v2f*)(bptr + kk);
#pragma unroll
      for (int s = 0; s < kSlabs; ++s) {
        const v2f a = *(const v2f*)(&h1s[(s * 16 + laneN) * kHStr + kk]);
        acc[s] = wmma_f32_k4(a, b, acc[s]);
      }
    }
#pragma unroll
    for (int s = 0; s < kSlabs; ++s)
#pragma unroll
      for (int v = 0; v < 8; ++v) {
        const int row = s * 16 + v + 8 * half;
        h2s[row * kHStr + wave * 16 + laneN] = acc[s][v];
      }
  }
  __syncthreads();

  // ------------------------------------------------------------------
  // 5) GN2 stats on h2, then fused residual + ReLU + head dot product.
  // ------------------------------------------------------------------
  {
    const int row = t & 63, chunk = t >> 6;
    float s = 0.0f, q = 0.0f;
    for (int i = 0; i < 32; ++i) {
      const float v = h2s[row * kHStr + chunk * 32 + i];
      s += v; q += v * v;
    }
    sA[chunk * 64 + row] = s;
    sB[chunk * 64 + row] = q;
  }
  __syncthreads();
  if (t < 64) {
    float s = 0.0f, q = 0.0f;
    for (int c = 0; c < 5; ++c) { s += sA[c * 64 + t]; q += sB[c * 64 + t]; }
    const float mu  = s * (1.0f / kHid);
    const float var = q * (1.0f / kHid) - mu * mu;
    rmu[t] = mu;
    rrs[t] = rsqrtf(var + kEps);
  }
  __syncthreads();
  {
    const int row = t & 63, chunk = t >> 6;
    const float mu = rmu[row], rs = rrs[row];
    float acc = 0.0f;
    for (int i = 0; i < 32; ++i) {
      const int c = chunk * 32 + i;
      float v = h2s[row * kHStr + c];
      v = (v - mu) * rs * g2w[c] + g2b[c] + hts[row * kHStr + c];
      v = fmaxf(v, 0.0f);
      acc += v * wh[c];
    }
    sA[chunk * 64 + row] = acc;
  }
  __syncthreads();
  if (t < 64) {
    float s = 0.0f;
    for (int c = 0; c < 5; ++c) s += sA[c * 64 + t];
    out[base + t] = s + bh[0];
  }
}

extern "C" void kernel_launch(void* const* d_in, const int* in_sizes, int n_in,
                              void* d_out, int out_size, void* d_ws, size_t ws_size,
                              hipStream_t stream) {
  const float* actors = (const float*)d_in[0];
  const float* paths  = (const float*)d_in[1];
  const float* Zact   = (const float*)d_in[2];
  const float* Zpat   = (const float*)d_in[3];
  const int*   u      = (const int*)d_in[4];
  const float* w1     = (const float*)d_in[5];
  const float* w2     = (const float*)d_in[6];
  const float* wt     = (const float*)d_in[7];
  const float* g1w    = (const float*)d_in[8];
  const float* g1b    = (const float*)d_in[9];
  const float* g2w    = (const float*)d_in[10];
  const float* g2b    = (const float*)d_in[11];
  const float* wh     = (const float*)d_in[12];
  const float* bh     = (const float*)d_in[13];
  float* out          = (float*)d_out;

  const int n_path = in_sizes[4];                 // 400000
  const int blocks = (n_path + kRows - 1) / kRows;

  // 166 KB dynamic LDS (WGP has 320 KB); opt in every call (stateless,
  // not a stream op -> safe under graph capture).
  (void)hipFuncSetAttribute((const void*)pathpred_kernel,
                            hipFuncAttributeMaxDynamicSharedMemorySize,
                            (int)kSmemBytes);

  pathpred_kernel<<<blocks, kThreads, kSmemBytes, stream>>>(
      actors, paths, Zact, Zpat, u, w1, w2, wt,
      g1w, g1b, g2w, g2b, wh, bh, out);
}